// DeepONet_13993003450991
// MI455X (gfx1250) — compile-verified
//
#include <hip/hip_runtime.h>
#include <hip/hip_bf16.h>
#include <math.h>

#define NN 50000
#define NE 1600000
#define ET (NE + NN)      // edges + self loops
#define NG 64

typedef float v2f __attribute__((ext_vector_type(2)));
typedef float v8f __attribute__((ext_vector_type(8)));

static inline unsigned gdiv(long n, int b) { return (unsigned)((n + b - 1) / b); }

// ---- order-preserving float<->uint encoding for atomicMax-based segment max ----
__device__ __forceinline__ unsigned enc_ord(float f) {
  unsigned b = __float_as_uint(f);
  return (b & 0x80000000u) ? ~b : (b | 0x80000000u);
}
__device__ __forceinline__ float dec_ord(unsigned u) {
  unsigned b = (u & 0x80000000u) ? (u ^ 0x80000000u) : ~u;
  return __uint_as_float(b);
}
// encoded(-inf) = ~bits(-inf)
#define ENC_NEG_INF 0x007FFFFFu

__device__ __forceinline__ void edge_sd(const int* __restrict__ ei, int e, int& s, int& d) {
  if (e < NE) { s = ei[e]; d = ei[NE + e]; }
  else        { s = e - NE; d = s; }           // appended self-loops
}

__device__ __forceinline__ float elu1(float v) { return v > 0.f ? v : (expf(v) - 1.f); }

// ---------------- utility fills (vectorized) ----------------
__global__ void k_fill4_f32(float4* __restrict__ p, float v, int n4) {
  int i = blockIdx.x * blockDim.x + threadIdx.x;
  if (i < n4) p[i] = make_float4(v, v, v, v);
}
__global__ void k_fill4_u32(uint4* __restrict__ p, unsigned v, int n4) {
  int i = blockIdx.x * blockDim.x + threadIdx.x;
  if (i < n4) p[i] = make_uint4(v, v, v, v);
}
__global__ void k_fill_f32(float* __restrict__ p, float v, int n) {
  int i = blockIdx.x * blockDim.x + threadIdx.x;
  if (i < n) p[i] = v;
}

// h1[n, 4q..4q+3] = x[n,0]*W[0,4q..] + x[n,1]*W[1,4q..]   (128 ch -> 32 quads)
__global__ void k_in_proj4(const float* __restrict__ x, const float* __restrict__ W,
                           float4* __restrict__ h, int n4) {
  int i = blockIdx.x * blockDim.x + threadIdx.x;
  if (i >= n4) return;
  int n = i >> 5, q = i & 31;
  float x0 = x[2 * n], x1 = x[2 * n + 1];
  float4 w0 = ((const float4*)W)[q];
  float4 w1 = ((const float4*)W)[32 + q];
  float4 r;
  r.x = x0 * w0.x + x1 * w1.x;
  r.y = x0 * w0.y + x1 * w1.y;
  r.z = x0 * w0.z + x1 * w1.z;
  r.w = x0 * w0.w + x1 * w1.w;
  h[i] = r;
}

// as_[n,h] = sum_c h[n,h,c]*a_src[h,c] ; ad_ likewise.  i = n*heads + h
__global__ void k_dots(const float* __restrict__ h, const float* __restrict__ a_src,
                       const float* __restrict__ a_dst, float* __restrict__ as_,
                       float* __restrict__ ad_, int heads) {
  int i = blockIdx.x * blockDim.x + threadIdx.x;
  if (i >= NN * heads) return;
  int hd = i % heads;
  const float4* hp  = (const float4*)(h + (size_t)i * 64);   // (n*heads+h)*64
  const float4* asp = (const float4*)(a_src + hd * 64);
  const float4* adp = (const float4*)(a_dst + hd * 64);
  float s1 = 0.f, s2 = 0.f;
#pragma unroll 4
  for (int c = 0; c < 16; ++c) {
    float4 v = hp[c], a = asp[c], b = adp[c];
    s1 += v.x * a.x + v.y * a.y + v.z * a.z + v.w * a.w;
    s2 += v.x * b.x + v.y * b.y + v.z * b.z + v.w * b.w;
  }
  as_[i] = s1; ad_[i] = s2;
}

// segment max of leaky_relu(as[src]+ad[dst]) grouped by dst
__global__ void k_edge_max(const int* __restrict__ ei, const float* __restrict__ as_,
                           const float* __restrict__ ad_, unsigned* __restrict__ m, int hshift) {
  int i = blockIdx.x * blockDim.x + threadIdx.x;
  int total = ET << hshift;
  if (i >= total) return;
  int e = i >> hshift, hd = i & ((1 << hshift) - 1), heads = 1 << hshift;
  int s, d; edge_sd(ei, e, s, d);
  float v = as_[s * heads + hd] + ad_[d * heads + hd];
  v = v > 0.f ? v : 0.2f * v;
  atomicMax(&m[d * heads + hd], enc_ord(v));
}

// ex = exp(e - m[dst]); store ex; denom[dst] += ex
__global__ void k_edge_exp(const int* __restrict__ ei, const float* __restrict__ as_,
                           const float* __restrict__ ad_, const unsigned* __restrict__ m,
                           float* __restrict__ ex, float* __restrict__ den, int hshift) {
  int i = blockIdx.x * blockDim.x + threadIdx.x;
  int total = ET << hshift;
  if (i >= total) return;
  int e = i >> hshift, hd = i & ((1 << hshift) - 1), heads = 1 << hshift;
  int s, d; edge_sd(ei, e, s, d);
  float v = as_[s * heads + hd] + ad_[d * heads + hd];
  v = v > 0.f ? v : 0.2f * v;
  float ev = expf(v - dec_ord(m[d * heads + hd]));
  ex[i] = ev;
  atomicAdd(&den[d * heads + hd], ev);
}

// out[dst, 4q..4q+3] += alpha * h[src, 4q..4q+3]
// i = e*(CH/4) + q, CH = heads*64, CH/4 = 1<<qshift (32 for H=2, 16 for H=1)
__global__ void k_edge_scatter4(const int* __restrict__ ei, const float* __restrict__ ex,
                                const float* __restrict__ den, const float* __restrict__ hsrc,
                                float* __restrict__ out, int heads, int qshift) {
  int i = blockIdx.x * blockDim.x + threadIdx.x;
  int total = ET << qshift;
  if (i >= total) return;
  int e = i >> qshift;
  int q = i & ((1 << qshift) - 1);
  int hd = q >> 4;                      // channel/64
  int s, d; edge_sd(ei, e, s, d);
  float alpha = ex[e * heads + hd] / (den[d * heads + hd] + 1e-16f);
  int CH = heads << 6;
  float4 hv = *(const float4*)(hsrc + (size_t)s * CH + 4 * q);
  float* op = out + (size_t)d * CH + 4 * q;
  atomicAdd(op + 0, alpha * hv.x);
  atomicAdd(op + 1, alpha * hv.y);
  atomicAdd(op + 2, alpha * hv.z);
  atomicAdd(op + 3, alpha * hv.w);
}

// out = elu(acc + b[c]) vectorized; qmask = (C/4)-1
__global__ void k_bias_elu4(const float4* __restrict__ acc, const float4* __restrict__ b,
                            float4* __restrict__ out, int n4, int qmask) {
  int i = blockIdx.x * blockDim.x + threadIdx.x;
  if (i >= n4) return;
  float4 a = acc[i];
  float4 bb = b[i & qmask];
  float4 r;
  r.x = elu1(a.x + bb.x);
  r.y = elu1(a.y + bb.y);
  r.z = elu1(a.z + bb.z);
  r.w = elu1(a.w + bb.w);
  out[i] = r;
}

// ---------------- WMMA GEMM: H[M,64] = X[M,128] @ W[128,64], M = 50000 = 3125*16 ----------------
// one wave per 16x16 output tile; V_WMMA_F32_16X16X4_F32, 32 k-steps of 4
__global__ void k_gemm_wmma(const float* __restrict__ X, const float* __restrict__ W,
                            float* __restrict__ H) {
  int tile = blockIdx.x;
  int tm = tile >> 2;            // 0..3124
  int tn = tile & 3;             // 0..3  (64 cols)
  int lane = threadIdx.x;        // wave32
  int half = lane >> 4;          // 0: K=k,k+1 ; 1: K=k+2,k+3
  int l16  = lane & 15;
  int m0 = tm * 16, n0 = tn * 16;
  v8f acc = {0.f, 0.f, 0.f, 0.f, 0.f, 0.f, 0.f, 0.f};
  const float* xrow = X + (size_t)(m0 + l16) * 128 + 2 * half;
  const float* wcol = W + n0 + l16 + (size_t)(2 * half) * 64;
#pragma unroll 4
  for (int k = 0; k < 128; k += 4) {
    // A 16x4: lanes 0-15 -> {K=k,k+1}, lanes 16-31 -> {K=k+2,k+3}, row = m0+l16
    v2f a; a.x = xrow[k]; a.y = xrow[k + 1];
    // B 4x16: VGPR0 rows {k, k+2}, VGPR1 rows {k+1, k+3}, col = n0+l16
    v2f b; b.x = wcol[(size_t)k * 64]; b.y = wcol[(size_t)(k + 1) * 64];
    acc = __builtin_amdgcn_wmma_f32_16x16x4_f32(false, a, false, b, (short)0, acc,
                                                false, false);
  }
  // C/D: VGPR r -> row m0+r (lanes 0-15) / m0+8+r (lanes 16-31), col n0+l16
#pragma unroll
  for (int r = 0; r < 8; ++r) {
    int m = m0 + r + half * 8;
    H[(size_t)m * 64 + n0 + l16] = acc[r];
  }
}

// ---------------- pooling + dense ----------------
// i = n*16 + q : sums[batch[n], 4q..4q+3] += x3[n, 4q..4q+3]
__global__ void k_pool4(const float* __restrict__ x3, const int* __restrict__ batch,
                        float* __restrict__ sums, float* __restrict__ cnt) {
  int i = blockIdx.x * blockDim.x + threadIdx.x;
  if (i >= NN * 16) return;
  int n = i >> 4, q = i & 15;
  int g = batch[n];
  float4 v = ((const float4*)x3)[i];
  float* sp = sums + g * 64 + 4 * q;
  atomicAdd(sp + 0, v.x);
  atomicAdd(sp + 1, v.y);
  atomicAdd(sp + 2, v.z);
  atomicAdd(sp + 3, v.w);
  if (q == 0) atomicAdd(&cnt[g], 1.f);
}
__global__ void k_pool_fin(const float* __restrict__ sums, const float* __restrict__ cnt,
                           float* __restrict__ pooled) {
  int i = blockIdx.x * blockDim.x + threadIdx.x;
  if (i >= NG * 64) return;
  pooled[i] = sums[i] / fmaxf(cnt[i >> 6], 1.f);
}
// Y[r,j] = act(sum_k X[r,k]*W[k,j] + b[j]) ; act: 0 none, 1 relu, 2 tanh
__global__ void k_dense(const float* __restrict__ X, const float* __restrict__ W,
                        const float* __restrict__ b, float* __restrict__ Y,
                        int rows, int K, int ncols, int act) {
  int i = blockIdx.x * blockDim.x + threadIdx.x;
  if (i >= rows * ncols) return;
  int r = i / ncols, j = i - r * ncols;
  float acc = b[j];
  for (int k = 0; k < K; ++k) acc += X[r * K + k] * W[k * ncols + j];
  if (act == 1) acc = fmaxf(acc, 0.f);
  else if (act == 2) acc = tanhf(acc);
  Y[i] = acc;
}
__global__ void k_mul(const float* __restrict__ a, const float* __restrict__ b,
                      float* __restrict__ c, int n) {
  int i = blockIdx.x * blockDim.x + threadIdx.x;
  if (i < n) c[i] = a[i] * b[i];
}

// ---------------- per-net driver ----------------
// P: W1,a_src1,a_dst1,b1,W2,a_src2,a_dst2,b2,mW1,mb1,mW2,mb2
static void run_net(const float* x, const int* ei, const int* batch, const float* const* P,
                    float* A, float* B, float* C, float* EX, float* AS, float* AD,
                    unsigned* M, float* DEN, float* SUMS, float* CNT, float* POOL,
                    float* HB, float* net_out, hipStream_t st) {
  // ---- GAT layer 1 (heads=2, C=64) ----
  k_fill4_u32<<<gdiv(NN / 2, 256), 256, 0, st>>>((uint4*)M, ENC_NEG_INF, NN / 2);
  k_fill4_f32<<<gdiv(NN / 2, 256), 256, 0, st>>>((float4*)DEN, 0.f, NN / 2);
  k_fill4_f32<<<gdiv(NN * 32, 256), 256, 0, st>>>((float4*)B, 0.f, NN * 32);
  k_in_proj4<<<gdiv(NN * 32, 256), 256, 0, st>>>(x, P[0], (float4*)A, NN * 32);
  k_dots<<<gdiv(NN * 2, 256), 256, 0, st>>>(A, P[1], P[2], AS, AD, 2);
  k_edge_max<<<gdiv((long)ET * 2, 256), 256, 0, st>>>(ei, AS, AD, M, 1);
  k_edge_exp<<<gdiv((long)ET * 2, 256), 256, 0, st>>>(ei, AS, AD, M, EX, DEN, 1);
  k_edge_scatter4<<<gdiv((long)ET * 32, 256), 256, 0, st>>>(ei, EX, DEN, A, B, 2, 5);
  k_bias_elu4<<<gdiv(NN * 32, 256), 256, 0, st>>>((const float4*)B, (const float4*)P[3],
                                                  (float4*)A, NN * 32, 31);   // x2 -> A
  // ---- GAT layer 2 (heads=1, C=64) ----
  k_fill4_u32<<<gdiv(NN / 4, 256), 256, 0, st>>>((uint4*)M, ENC_NEG_INF, NN / 4);
  k_fill4_f32<<<gdiv(NN / 4, 256), 256, 0, st>>>((float4*)DEN, 0.f, NN / 4);
  k_fill4_f32<<<gdiv(NN * 16, 256), 256, 0, st>>>((float4*)C, 0.f, NN * 16);
  k_gemm_wmma<<<3125 * 4, 32, 0, st>>>(A, P[4], B);                 // h2 -> B
  k_dots<<<gdiv(NN, 256), 256, 0, st>>>(B, P[5], P[6], AS, AD, 1);
  k_edge_max<<<gdiv((long)ET, 256), 256, 0, st>>>(ei, AS, AD, M, 0);
  k_edge_exp<<<gdiv((long)ET, 256), 256, 0, st>>>(ei, AS, AD, M, EX, DEN, 0);
  k_edge_scatter4<<<gdiv((long)ET * 16, 256), 256, 0, st>>>(ei, EX, DEN, B, C, 1, 4);
  k_bias_elu4<<<gdiv(NN * 16, 256), 256, 0, st>>>((const float4*)C, (const float4*)P[7],
                                                  (float4*)C, NN * 16, 15);   // x3 -> C
  // ---- mean pool + MLP ----
  k_fill4_f32<<<gdiv(NG * 16, 256), 256, 0, st>>>((float4*)SUMS, 0.f, NG * 16);
  k_fill_f32<<<1, 64, 0, st>>>(CNT, 0.f, NG);
  k_pool4<<<gdiv(NN * 16, 256), 256, 0, st>>>(C, batch, SUMS, CNT);
  k_pool_fin<<<gdiv(NG * 64, 256), 256, 0, st>>>(SUMS, CNT, POOL);
  k_dense<<<gdiv(NG * 64, 256), 256, 0, st>>>(POOL, P[8], P[9], HB, NG, 64, 64, 1);
  k_dense<<<gdiv(NG * 64, 256), 256, 0, st>>>(HB, P[10], P[11], net_out, NG, 64, 64, 0);
}

extern "C" void kernel_launch(void* const* d_in, const int* in_sizes, int n_in,
                              void* d_out, int out_size, void* d_ws, size_t ws_size,
                              hipStream_t stream) {
  const float* x_branch = (const float*)d_in[0];
  const float* x_trunk  = (const float*)d_in[1];
  const int*   ei       = (const int*)d_in[2];   // [2, NE] row-major: src then dst
  const int*   batch    = (const int*)d_in[3];
  const float* bp[12]; for (int i = 0; i < 12; ++i) bp[i] = (const float*)d_in[4 + i];
  const float* tp[12]; for (int i = 0; i < 12; ++i) tp[i] = (const float*)d_in[16 + i];
  const float* hW1 = (const float*)d_in[28];
  const float* hb1 = (const float*)d_in[29];
  const float* hW2 = (const float*)d_in[30];
  const float* hb2 = (const float*)d_in[31];

  // ---- workspace layout (floats; every region size is a multiple of 4 -> 16B aligned) ----
  float* ws = (float*)d_ws;
  size_t o = 0;
  float*    A    = ws + o; o += (size_t)NN * 128;   // h1 / x2
  float*    B    = ws + o; o += (size_t)NN * 128;   // out1 / h2
  float*    C    = ws + o; o += (size_t)NN * 64;    // out2 / x3
  float*    EX   = ws + o; o += (size_t)ET * 2;     // per-edge exp weights
  float*    AS   = ws + o; o += (size_t)NN * 2;
  float*    AD   = ws + o; o += (size_t)NN * 2;
  unsigned* M    = (unsigned*)(ws + o); o += (size_t)NN * 2;
  float*    DEN  = ws + o; o += (size_t)NN * 2;
  float*    SUMS = ws + o; o += NG * 64;
  float*    CNT  = ws + o; o += NG;
  float*    POOL = ws + o; o += NG * 64;
  float*    HB   = ws + o; o += NG * 64;
  float*    OUTB = ws + o; o += NG * 64;
  float*    OUTT = ws + o; o += NG * 64;
  (void)in_sizes; (void)n_in; (void)out_size; (void)ws_size;

  run_net(x_branch, ei, batch, bp, A, B, C, EX, AS, AD, M, DEN, SUMS, CNT, POOL, HB, OUTB, stream);
  run_net(x_trunk,  ei, batch, tp, A, B, C, EX, AS, AD, M, DEN, SUMS, CNT, POOL, HB, OUTT, stream);

  // head: tanh(relu((b*t)@W1+b1)@W2+b2) -> [64,2]
  k_mul<<<gdiv(NG * 64, 256), 256, 0, stream>>>(OUTB, OUTT, POOL, NG * 64);
  k_dense<<<gdiv(NG * 64, 256), 256, 0, stream>>>(POOL, hW1, hb1, HB, NG, 64, 64, 1);
  k_dense<<<1, 256, 0, stream>>>(HB, hW2, hb2, (float*)d_out, NG, 64, 2, 2);
}